// CorrNeigh_37958920962306
// MI455X (gfx1250) — compile-verified
//
#include <hip/hip_runtime.h>

typedef __attribute__((ext_vector_type(2))) float v2f;
typedef __attribute__((ext_vector_type(4))) float v4f;
typedef __attribute__((ext_vector_type(8))) float v8f;

#define HH   256
#define WW   256
#define CCH  128
#define HWSZ (HH * WW)
#define NK   49
#define LROW 836   // LDS row stride in floats (not a multiple of 64 banks)

// One wave computes one 4x4 block of x-pixels for one batch image.
// N (B-operand) = 16 x-pixels (4x4), K = channels (steps of 4),
// M (A-operand) = 16 y-pixels (4x4) at block offsets dy,dx in {-3,1,5}.
// D[m,n] = sum_c y[c,pix_m] * x[c,pix_n] -> out[b, delta, pix_n].
// The 9 aligned y-blocks give every delta in [-3,3]^2 exactly once per pixel.
// The 8 waves of a block cover a 4x32 strip so output stores coalesce.
__global__ __launch_bounds__(256) void corr7x7_wmma_f32(
    const float* __restrict__ x, const float* __restrict__ y,
    float* __restrict__ out)
{
  __shared__ __align__(16) float lds[8][LROW];

  const int tid  = threadIdx.x;
  const int lane = tid & 31;
  const int wave = tid >> 5;

  // block -> (b, tile row, 32-px strip); wave -> one 4x4 tile in the strip
  int strip = blockIdx.x & 7;
  int th    = (blockIdx.x >> 3) & 63;
  int b     = blockIdx.x >> 9;          // 8 * 64 * 8 = 4096 blocks
  int h0  = th << 2;
  int w0s = strip << 5;
  int w0  = w0s + ((tid >> 5) << 2);

  // Fragment lane mapping (ISA 7.12.2): lanes 0-15 hold K=kb,kb+1; lanes
  // 16-31 hold K=kb+2,kb+3 of the same 16 M/N entries.
  const int n  = lane & 15;
  const int nh = n >> 2, nw = n & 3;
  const int kb = (lane >> 4) << 1;

  const float* xb = x + (size_t)b * CCH * HWSZ;
  const float* yb = y + (size_t)b * CCH * HWSZ;

  const int xoff = kb * HWSZ + (h0 + nh) * WW + (w0 + nw);

  int  aoff[9];
  bool aval[9];
#pragma unroll
  for (int t = 0; t < 9; ++t) {
    int dy = (t / 3) * 4 - 3;
    int dx = (t % 3) * 4 - 3;
    int yy = h0 + dy + nh;
    int xx = w0 + dx + nw;
    bool v = ((unsigned)yy < HH) && ((unsigned)xx < WW);
    aval[t] = v;
    aoff[t] = v ? (kb * HWSZ + yy * WW + xx) : 0;
  }

  v8f acc[9];
#pragma unroll
  for (int t = 0; t < 9; ++t) acc[t] = (v8f){};

  // interior tiles touch no padding: y rows [h0-3, h0+8] all in range
  const bool interior = (h0 >= 4) && (h0 <= HH - 12) &&
                        (w0 >= 4) && (w0 <= WW - 12);

  auto loadI = [&](int c, v2f& bf, v2f (&af)[9]) {  // no masking
    int cs = c * HWSZ;
    bf.x = xb[cs + xoff];
    bf.y = xb[cs + xoff + HWSZ];
#pragma unroll
    for (int t = 0; t < 9; ++t) {
      af[t].x = yb[cs + aoff[t]];
      af[t].y = yb[cs + aoff[t] + HWSZ];
    }
  };
  auto loadB = [&](int c, v2f& bf, v2f (&af)[9]) {  // branchless select
    int cs = c * HWSZ;
    bf.x = xb[cs + xoff];
    bf.y = xb[cs + xoff + HWSZ];
#pragma unroll
    for (int t = 0; t < 9; ++t) {
      float v0 = yb[cs + aoff[t]];
      float v1 = yb[cs + aoff[t] + HWSZ];
      af[t].x = aval[t] ? v0 : 0.0f;
      af[t].y = aval[t] ? v1 : 0.0f;
    }
  };
  auto wmma9 = [&](v2f (&af)[9], v2f bf) {
#pragma unroll
    for (int t = 0; t < 9; ++t) {
      acc[t] = __builtin_amdgcn_wmma_f32_16x16x4_f32(
          false, af[t], false, bf, (short)0, acc[t], false, false);
    }
  };

  // Explicit two-stage ping-pong (no fragment copies).
  v2f b0, b1, a0[9], a1[9];
  if (interior) {
    loadI(0, b0, a0);
    loadI(4, b1, a1);
    for (int c = 0; c < CCH; c += 8) {
      wmma9(a0, b0);
      if (c + 8 < CCH) loadI(c + 8, b0, a0);
      wmma9(a1, b1);
      if (c + 12 < CCH) loadI(c + 12, b1, a1);
    }
  } else {
    loadB(0, b0, a0);
    loadB(4, b1, a1);
    for (int c = 0; c < CCH; c += 8) {
      wmma9(a0, b0);
      if (c + 8 < CCH) loadB(c + 8, b0, a0);
      wmma9(a1, b1);
      if (c + 12 < CCH) loadB(c + 12, b1, a1);
    }
  }

  // Stage results to LDS. Out-of-window (m,n) pairs go to a per-lane dump
  // slot -> every ds_store is unconditional (no exec masking).
  const int mofs = (lane >> 4) << 3;    // lanes 16-31 hold M = r + 8
#pragma unroll
  for (int t = 0; t < 9; ++t) {
    int dy = (t / 3) * 4 - 3;
    int dx = (t % 3) * 4 - 3;
#pragma unroll
    for (int r = 0; r < 8; ++r) {
      int m  = r + mofs;
      int di = dy + (m >> 2) - nh + 3;
      int dj = dx + (m & 3) - nw + 3;
      bool ok  = ((unsigned)di < 7u) && ((unsigned)dj < 7u);
      int addr = ok ? ((di * 7 + dj) * 16 + n) : (800 + lane);
      lds[wave][addr] = acc[t][r];
    }
  }
  __syncthreads();

  // Coalesced writeback: id = (k*4 + row)*8 + seg; lanes 0-7 cover 128
  // contiguous bytes of out[b][k][h0+row][w0s .. w0s+31].
  float* ob = out + (size_t)b * NK * HWSZ + (size_t)h0 * WW + w0s;
#pragma unroll
  for (int i = 0; i < 6; ++i) {
    int id  = i * 256 + tid;            // 6*256 + 32 = 1568 = 49*4*8 total
    int seg = id & 7, row = (id >> 3) & 3, k = id >> 5;
    v4f v = *(const v4f*)&lds[seg][k * 16 + row * 4];
    __builtin_nontemporal_store(v, (v4f*)&ob[k * HWSZ + row * WW + seg * 4]);
  }
  if (tid < 32) {
    int id  = 1536 + tid;
    int seg = id & 7, row = (id >> 3) & 3, k = id >> 5;
    v4f v = *(const v4f*)&lds[seg][k * 16 + row * 4];
    __builtin_nontemporal_store(v, (v4f*)&ob[k * HWSZ + row * WW + seg * 4]);
  }
}

extern "C" void kernel_launch(void* const* d_in, const int* in_sizes, int n_in,
                              void* d_out, int out_size, void* d_ws, size_t ws_size,
                              hipStream_t stream) {
  const float* x = (const float*)d_in[0];
  const float* y = (const float*)d_in[1];
  float* out     = (float*)d_out;
  (void)in_sizes; (void)n_in; (void)out_size; (void)d_ws; (void)ws_size;

  dim3 grid(4096), block(256);
  hipLaunchKernelGGL(corr7x7_wmma_f32, grid, block, 0, stream, x, y, out);
}